// GAT_50448685859069
// MI455X (gfx1250) — compile-verified
//
#include <hip/hip_runtime.h>
#include <hip/hip_bf16.h>

typedef __attribute__((ext_vector_type(2))) float v2f;
typedef __attribute__((ext_vector_type(8))) float v8f;

#define NEG_SLOPE 0.2f

// ---------------------------------------------------------------------------
// utility
// ---------------------------------------------------------------------------
__global__ void fill_f32(float* __restrict__ p, float v, int n) {
    int i = blockIdx.x * blockDim.x + threadIdx.x;
    if (i < n) p[i] = v;
}

__device__ __forceinline__ float atomicMaxF(float* addr, float val) {
    // sign-split trick -> native global_atomic_max_i32 / global_atomic_min_u32
    if (val >= 0.0f)
        return __int_as_float(atomicMax((int*)addr, __float_as_int(val)));
    return __uint_as_float(atomicMin((unsigned int*)addr, __float_as_uint(val)));
}

// Butterfly sum over each 16-lane half of a wave32 via ds_swizzle_b32
// group-of-32 XOR mode: offset = (xor_mask<<10) | (or_mask<<5) | and_mask.
// Replaces __shfl_xor's ds_bpermute + cmp/cndmask sequence.
__device__ __forceinline__ float red16_sum(float v) {
    v += __int_as_float(__builtin_amdgcn_ds_swizzle(__float_as_int(v), (8 << 10) | 0x1f));
    v += __int_as_float(__builtin_amdgcn_ds_swizzle(__float_as_int(v), (4 << 10) | 0x1f));
    v += __int_as_float(__builtin_amdgcn_ds_swizzle(__float_as_int(v), (2 << 10) | 0x1f));
    v += __int_as_float(__builtin_amdgcn_ds_swizzle(__float_as_int(v), (1 << 10) | 0x1f));
    return v;
}

// ---------------------------------------------------------------------------
// Layer-1 GEMM: feat1[N,128] = X[N,128] @ W1[128,128], fused el/er epilogue.
// Block = 8 waves; wave w owns head w (16-col tile, D=16) for 16 rows.
// V_WMMA_F32_16X16X4_F32: exact fp32 matrix op (GEMM is not the bottleneck;
// no reason to downconvert).
// ---------------------------------------------------------------------------
__global__ __launch_bounds__(256) void gat_gemm1(
    const float* __restrict__ X, const float* __restrict__ W,
    const float* __restrict__ al, const float* __restrict__ ar,
    float* __restrict__ feat, float* __restrict__ el, float* __restrict__ er,
    int N)
{
    const int wave = threadIdx.x >> 5;      // 0..7 : head / 16-col tile
    const int lane = threadIdx.x & 31;
    const int n    = lane & 15;             // col within tile / A-row
    const int hi   = lane >> 4;             // half-wave select
    const int row0 = blockIdx.x * 16;
    const int col0 = wave * 16;

    const int rA = min(row0 + n, N - 1);
    const float* __restrict__ xrow = X + (size_t)rA * 128;

    v8f c = {};
    #pragma unroll 4
    for (int k = 0; k < 128; k += 4) {
        v2f a, b;
        // A 16x4: lane m=n, VGPR pair holds K = 2*hi, 2*hi+1
        a.x = xrow[k + 2 * hi];
        a.y = xrow[k + 2 * hi + 1];
        // B 4x16: lane n, VGPR pair holds K = 2*hi, 2*hi+1
        b.x = W[(size_t)(k + 2 * hi) * 128 + col0 + n];
        b.y = W[(size_t)(k + 2 * hi + 1) * 128 + col0 + n];
        c = __builtin_amdgcn_wmma_f32_16x16x4_f32(
                false, a, false, b, (short)0, c, false, false);
    }

    const float wl = al[col0 + n];          // attn_l1[head=wave][n]
    const float wr = ar[col0 + n];
    #pragma unroll
    for (int r = 0; r < 8; ++r) {
        const int row = row0 + r + hi * 8;  // C layout: lanes 16-31 hold M=r+8
        float cv = c[r];
        if (row < N) feat[(size_t)row * 128 + col0 + n] = cv;
        float pl = red16_sum(cv * wl);
        float pr = red16_sum(cv * wr);
        if (n == 0 && row < N) {
            el[(size_t)row * 8 + wave] = pl;
            er[(size_t)row * 8 + wave] = pr;
        }
    }
}

// ---------------------------------------------------------------------------
// Layer-2 GEMM: feat2[N,64] = H[N,128] @ W2[128,64], fused el2/er2 partials.
// Block = 8 waves: 2 row-tiles x 4 col-tiles (32 rows per block).
// ---------------------------------------------------------------------------
__global__ __launch_bounds__(256) void gat_gemm2(
    const float* __restrict__ Hm, const float* __restrict__ W,
    const float* __restrict__ al, const float* __restrict__ ar,
    float* __restrict__ feat, float* __restrict__ el, float* __restrict__ er,
    int N)
{
    const int wave = threadIdx.x >> 5;
    const int rt   = wave >> 2;
    const int ct   = wave & 3;
    const int lane = threadIdx.x & 31;
    const int n    = lane & 15;
    const int hi   = lane >> 4;
    const int row0 = blockIdx.x * 32 + rt * 16;
    const int col0 = ct * 16;

    const int rA = min(row0 + n, N - 1);
    const float* __restrict__ hrow = Hm + (size_t)rA * 128;

    v8f c = {};
    #pragma unroll 4
    for (int k = 0; k < 128; k += 4) {
        v2f a, b;
        a.x = hrow[k + 2 * hi];
        a.y = hrow[k + 2 * hi + 1];
        b.x = W[(size_t)(k + 2 * hi) * 64 + col0 + n];
        b.y = W[(size_t)(k + 2 * hi + 1) * 64 + col0 + n];
        c = __builtin_amdgcn_wmma_f32_16x16x4_f32(
                false, a, false, b, (short)0, c, false, false);
    }

    const float wl = al[col0 + n];
    const float wr = ar[col0 + n];
    #pragma unroll
    for (int r = 0; r < 8; ++r) {
        const int row = row0 + r + hi * 8;
        float cv = c[r];
        if (row < N) feat[(size_t)row * 64 + col0 + n] = cv;
        float pl = red16_sum(cv * wl);
        float pr = red16_sum(cv * wr);
        if (n == 0 && row < N) {            // 4 col-tile waves sum the full dot
            atomicAdd(&el[row], pl);
            atomicAdd(&er[row], pr);
        }
    }
}

// ---------------------------------------------------------------------------
// Edge softmax pass 1: segmented max (templated H -> b128-vectorized loads)
// ---------------------------------------------------------------------------
template <int H>
__global__ __launch_bounds__(256) void edge_max(
    const int* __restrict__ src, const int* __restrict__ dst,
    const float* __restrict__ el, const float* __restrict__ er,
    float* __restrict__ m, int E)
{
    int e = blockIdx.x * blockDim.x + threadIdx.x;
    if (e >= E) return;
    const int s = src[e], d = dst[e];
    #pragma unroll
    for (int h = 0; h < H; ++h) {
        float v = el[(size_t)s * H + h] + er[(size_t)d * H + h];
        v = v > 0.0f ? v : NEG_SLOPE * v;
        atomicMaxF(&m[(size_t)d * H + h], v);
    }
}

// ---------------------------------------------------------------------------
// Edge softmax pass 2: segmented sum of exp(e - m[dst])
// ---------------------------------------------------------------------------
template <int H>
__global__ __launch_bounds__(256) void edge_sum(
    const int* __restrict__ src, const int* __restrict__ dst,
    const float* __restrict__ el, const float* __restrict__ er,
    const float* __restrict__ m, float* __restrict__ ssum, int E)
{
    int e = blockIdx.x * blockDim.x + threadIdx.x;
    if (e >= E) return;
    const int s = src[e], d = dst[e];
    #pragma unroll
    for (int h = 0; h < H; ++h) {
        float v = el[(size_t)s * H + h] + er[(size_t)d * H + h];
        v = v > 0.0f ? v : NEG_SLOPE * v;
        float p = __expf(v - m[(size_t)d * H + h]);
        atomicAdd(&ssum[(size_t)d * H + h], p);
    }
}

// ---------------------------------------------------------------------------
// Edge aggregate: out[dst, col..col+3] += a(edge,head) * feat[src, col..col+3]
// One thread per (edge, group of 4 columns): amortizes the attention-weight
// recompute 4x, loads feat as float4 (global_load_b128), scatters with native
// global_atomic_add_f32 (L2-resident: feat + accumulator both fit in 192MB L2).
// C = H*D columns; logG = log2(C/4), logD = log2(D).
// ---------------------------------------------------------------------------
template <int LOGG, int LOGD>
__global__ __launch_bounds__(256) void edge_aggregate4(
    const int* __restrict__ src, const int* __restrict__ dst,
    const float* __restrict__ el, const float* __restrict__ er,
    const float* __restrict__ m, const float* __restrict__ ssum,
    const float* __restrict__ feat, float* __restrict__ out,
    long long nthreads)
{
    long long t = (long long)blockIdx.x * blockDim.x + threadIdx.x;
    if (t >= nthreads) return;
    constexpr int G = 1 << LOGG;            // groups of 4 columns
    constexpr int C = G * 4;                // total columns
    constexpr int H = C >> LOGD;            // heads
    const int g   = (int)(t & (long long)(G - 1));
    const long long e = t >> LOGG;
    const int col = g * 4;
    const int h   = col >> LOGD;
    const int s = src[e], d = dst[e];

    float v = el[(size_t)s * H + h] + er[(size_t)d * H + h];
    v = v > 0.0f ? v : NEG_SLOPE * v;
    float p = __expf(v - m[(size_t)d * H + h]);
    float a = p / fmaxf(ssum[(size_t)d * H + h], 1e-9f);

    const float4 f = *(const float4*)(feat + (size_t)s * C + col);
    float* o = out + (size_t)d * C + col;
    atomicAdd(o + 0, a * f.x);
    atomicAdd(o + 1, a * f.y);
    atomicAdd(o + 2, a * f.z);
    atomicAdd(o + 3, a * f.w);
}

// ---------------------------------------------------------------------------
// bias + ReLU (in-place), C = 128
// ---------------------------------------------------------------------------
__global__ void bias_relu_128(float* __restrict__ h, const float* __restrict__ bias, int n)
{
    int i = blockIdx.x * blockDim.x + threadIdx.x;
    if (i >= n) return;
    float v = h[i] + bias[i & 127];
    h[i] = v > 0.0f ? v : 0.0f;
}

// final bias add on d_out, C = 64 (mean over 1 head == identity)
__global__ void bias_out_64(float* __restrict__ out, const float* __restrict__ bias, int n)
{
    int i = blockIdx.x * blockDim.x + threadIdx.x;
    if (i >= n) return;
    out[i] = out[i] + bias[i & 63];
}

// ---------------------------------------------------------------------------
// launch
// ---------------------------------------------------------------------------
extern "C" void kernel_launch(void* const* d_in, const int* in_sizes, int n_in,
                              void* d_out, int out_size, void* d_ws, size_t ws_size,
                              hipStream_t stream) {
    const float* x   = (const float*)d_in[0];
    const int*   src = (const int*)d_in[1];
    const int*   dst = (const int*)d_in[2];
    const float* W1  = (const float*)d_in[3];
    const float* al1 = (const float*)d_in[4];
    const float* ar1 = (const float*)d_in[5];
    const float* b1  = (const float*)d_in[6];
    const float* W2  = (const float*)d_in[7];
    const float* al2 = (const float*)d_in[8];
    const float* ar2 = (const float*)d_in[9];
    const float* b2  = (const float*)d_in[10];
    float* out = (float*)d_out;

    const int N = in_sizes[0] / 128;
    const int E = in_sizes[1];

    // workspace layout (floats), ~143 MB total
    float* ws    = (float*)d_ws;
    float* feat1 = ws;                       // N*128
    float* el1   = feat1 + (size_t)N * 128;  // N*8
    float* er1   = el1 + (size_t)N * 8;
    float* m1    = er1 + (size_t)N * 8;
    float* s1    = m1 + (size_t)N * 8;
    float* agg1  = s1 + (size_t)N * 8;       // N*128 (becomes h after bias+relu)
    float* feat2 = agg1 + (size_t)N * 128;   // N*64
    float* el2   = feat2 + (size_t)N * 64;   // N
    float* er2   = el2 + N;
    float* m2    = er2 + N;
    float* s2    = m2 + N;

    const int B = 256;
    auto cdiv = [](long long a, long long b) { return (int)((a + b - 1) / b); };

    // ---- init ----
    fill_f32<<<cdiv((long long)N * 8, B), B, 0, stream>>>(m1, -INFINITY, N * 8);
    fill_f32<<<cdiv((long long)N * 8, B), B, 0, stream>>>(s1, 0.0f, N * 8);
    fill_f32<<<cdiv((long long)N * 128, B), B, 0, stream>>>(agg1, 0.0f, N * 128);
    fill_f32<<<cdiv(N, B), B, 0, stream>>>(m2, -INFINITY, N);
    fill_f32<<<cdiv(N, B), B, 0, stream>>>(s2, 0.0f, N);
    fill_f32<<<cdiv(N, B), B, 0, stream>>>(el2, 0.0f, N);
    fill_f32<<<cdiv(N, B), B, 0, stream>>>(er2, 0.0f, N);
    fill_f32<<<cdiv((long long)N * 64, B), B, 0, stream>>>(out, 0.0f, N * 64);

    // ---- layer 1 ----
    gat_gemm1<<<cdiv(N, 16), 256, 0, stream>>>(x, W1, al1, ar1, feat1, el1, er1, N);
    edge_max<8><<<cdiv(E, B), B, 0, stream>>>(src, dst, el1, er1, m1, E);
    edge_sum<8><<<cdiv(E, B), B, 0, stream>>>(src, dst, el1, er1, m1, s1, E);
    {
        long long nt = (long long)E * 32;    // 32 groups of 4 cols (C=128)
        edge_aggregate4<5, 4><<<cdiv(nt, B), B, 0, stream>>>(
            src, dst, el1, er1, m1, s1, feat1, agg1, nt);
    }
    bias_relu_128<<<cdiv((long long)N * 128, B), B, 0, stream>>>(agg1, b1, N * 128);

    // ---- layer 2 ----
    gat_gemm2<<<cdiv(N, 32), 256, 0, stream>>>(agg1, W2, al2, ar2, feat2, el2, er2, N);
    edge_max<1><<<cdiv(E, B), B, 0, stream>>>(src, dst, el2, er2, m2, E);
    edge_sum<1><<<cdiv(E, B), B, 0, stream>>>(src, dst, el2, er2, m2, s2, E);
    {
        long long nt = (long long)E * 16;    // 16 groups of 4 cols (C=64)
        edge_aggregate4<4, 6><<<cdiv(nt, B), B, 0, stream>>>(
            src, dst, el2, er2, m2, s2, feat2, out, nt);
    }
    bias_out_64<<<cdiv((long long)N * 64, B), B, 0, stream>>>(out, b2, N * 64);
}